// AttentionConvolution_51994874085870
// MI455X (gfx1250) — compile-verified
//
#include <hip/hip_runtime.h>
#include <math.h>

// ---------------------------------------------------------------------------
// AttentionConvolution on MI455X (gfx1250), fp32 WMMA (V_WMMA_F32_16X16X4_F32)
//
// Fusion: the last radial-MLP layer (100 -> 1024) is contracted with
// features[b] into per-b effective matrices Wk_eff[b], Wv_eff[b] (100x32),
// cutting total work from ~15 GMAC to ~1.8 GMAC.  All GEMMs run on the fp32
// WMMA pipe.  All B operands are pre-packed into WMMA fragment order so the
// inner loops issue one unconditional coalesced global_load_b64 per K-step
// (no exec-mask guards, zero padding baked in at pack time).  Transcendental
// epilogues use the hardware v_rcp_f32 instead of IEEE divide expansions.
// ---------------------------------------------------------------------------

typedef float v2f __attribute__((ext_vector_type(2)));
typedef float v8f __attribute__((ext_vector_type(8)));

#define NPTS    256
#define CCH     32
#define HID     100      // NB_KEY == H_KEY == H_VAL == 100
#define HPAD    112      // 7 tiles of 16
#define NTILES  7
#define KSTEPS  25       // K = 100 = 25 * 4
#define WPACK   (NTILES * KSTEPS * 64)   // 11200 floats per packed 100x100 W

#define STEP_K  (10.0f / 99.0f)   // linspace(0,10,100) step
#define STEP_V  5.0f              // linspace(0,10,3)  step
#define PI_2    1.57079632679489662f
#define RSQRT3  0.57735026918962576f

__device__ __forceinline__ v8f wmma_f32(v2f a, v2f b, v8f c) {
  // D = A(16x4) * B(4x16) + C(16x16), fp32, wave32
  return __builtin_amdgcn_wmma_f32_16x16x4_f32(false, a, false, b, (short)0, c,
                                               false, false);
}

__device__ __forceinline__ float swishf(float z) {
  // z * sigmoid(z) with hardware v_rcp_f32 (no IEEE divide expansion)
  return z * __builtin_amdgcn_rcpf(1.0f + __expf(-z));
}

__device__ __forceinline__ float cos_bump(float d, float center, float step) {
  float x = (d - center) * (PI_2 / step);
  float c = __cosf(x);
  return (fabsf(x) < PI_2) ? c * c : 0.0f;
}

// Packed B-fragment offset for element (k,n) of a KxN weight used at tile
// granularity 16: lane = 16*((k>>1)&1) + (n&15), pair slot = k&1.
__device__ __forceinline__ int bpack_off(int k, int n) {
  const int kk    = k >> 2;
  const int half  = (k >> 1) & 1;
  const int p     = k & 1;
  const int ntile = n >> 4;
  const int lane  = (half << 4) | (n & 15);
  return (((ntile * KSTEPS + kk) << 5) + lane) * 2 + p;
}

// ---------------------------------------------------------------------------
// Kernel 0: pack kW1 / kW2 (100x100) into WMMA-B fragment order, zero-padded
// to 112 columns.  11200 floats each.  grid = 88 x 256.
// ---------------------------------------------------------------------------
__global__ void pack_weights_kernel(const float* __restrict__ kW1,
                                    const float* __restrict__ kW2,
                                    float* __restrict__ kW1p,
                                    float* __restrict__ kW2p) {
  const int flat = blockIdx.x * 256 + threadIdx.x;
  if (flat >= 2 * WPACK) return;
  const int which = flat / WPACK;
  const int rem   = flat % WPACK;
  const int t    = rem / (KSTEPS * 64);
  const int r2   = rem % (KSTEPS * 64);
  const int kk   = r2 >> 6;
  const int lane = (r2 >> 1) & 31;
  const int p    = r2 & 1;
  const int k = 4 * kk + 2 * (lane >> 4) + p;       // 0..99
  const int n = t * 16 + (lane & 15);               // 0..111
  const float* W = which ? kW2 : kW1;
  float* Wp      = which ? kW2p : kW1p;
  Wp[rem] = (n < HID) ? W[k * HID + n] : 0.0f;
}

// ---------------------------------------------------------------------------
// Kernel 1: per-b precompute.
//   query[b,:]    = features[b,:] @ Wq
//   Wk_eff[b][h,i] = 0.1 * sum_j kW3[h, i*32+j] * f[b,j]   (packed B order)
//   Wv_eff[b][h,i] = 0.1 * sum_j vW2[h, i*32+j] * f[b,j]   (packed B order)
// grid = 256 blocks (one per b), block = 128
// ---------------------------------------------------------------------------
__global__ void prep_kernel(const float* __restrict__ features,
                            const float* __restrict__ Wq,
                            const float* __restrict__ kW3,
                            const float* __restrict__ vW2,
                            float* __restrict__ query,
                            float* __restrict__ Wk_eff,
                            float* __restrict__ Wv_eff) {
  __shared__ float f[CCH];
  const int b = blockIdx.x;
  const int tid = threadIdx.x;
  if (tid < CCH) f[tid] = features[b * CCH + tid];
  __syncthreads();

  if (tid < CCH) {
    float acc = 0.0f;
    for (int j = 0; j < CCH; ++j) acc += f[j] * Wq[j * CCH + tid];
    query[b * CCH + tid] = acc;
  }

  for (int idx = tid; idx < HID * CCH; idx += 128) {
    const int h = idx >> 5;        // 0..99  (k of the packed B)
    const int i = idx & 31;        // 0..31  (n of the packed B)
    const float* k3 = kW3 + (size_t)h * (CCH * CCH) + i * CCH;
    const float* v2 = vW2 + (size_t)h * (CCH * CCH) + i * CCH;
    float ak = 0.0f, av = 0.0f;
    for (int j = 0; j < CCH; ++j) {
      ak += k3[j] * f[j];
      av += v2[j] * f[j];
    }
    const int poff = bpack_off(h, i);               // 0..3199
    Wk_eff[(size_t)b * (HID * CCH) + poff] = ak * 0.1f;  // /sqrt(100)
    Wv_eff[(size_t)b * (HID * CCH) + poff] = av * 0.1f;
  }
}

// ---------------------------------------------------------------------------
// Kernel 2: per-pair hidden activations (the bulk of the FLOPs).
//   row r = b*256 + a  (all 32 rows of a WG share b)
//   H2[r,:] = swish( swish(basis_k(d) @ kW1 /10) @ kW2 /10 )   [100, pad 112]
//   Hv[r,:] = swish( basis_v(d) @ vW1 / sqrt(3) )              [100, pad 112]
// 32 rows per WG, 8 waves: wave w -> M-tile (w>>2), N-tiles {w&3, (w&3)+4}.
// A tiles in LDS (b64 reads), B from packed global (coalesced b64, L2-hot).
// grid = 2048, block = 256
// ---------------------------------------------------------------------------
__global__ void hidden_kernel(const float* __restrict__ geom,
                              const float* __restrict__ kW1p,
                              const float* __restrict__ kW2p,
                              const float* __restrict__ vW1,
                              float* __restrict__ H2,
                              float* __restrict__ Hv) {
  __shared__ float sA[32][HPAD];   // basis tile (GEMM1 A)
  __shared__ float sH[32][HPAD];   // hidden1 tile (GEMM2 A)
  __shared__ float sd[32];

  const int tid  = threadIdx.x;
  const int lane = tid & 31;
  const int wave = tid >> 5;
  const int half = lane >> 4;      // K-pair select (ISA A layout)
  const int l16  = lane & 15;      // M index within tile
  const int mt   = wave >> 2;      // M-tile (0/1)
  const int r0   = blockIdx.x * 32;
  const int mrow = mt * 16 + l16;  // A row this lane reads

  __builtin_prefetch(kW1p, 0, 1);  // global_prefetch_b8 (L2 warm)
  __builtin_prefetch(kW2p, 0, 1);

  // pairwise distances for the 32 rows of this WG
  if (tid < 32) {
    const int r = r0 + tid;
    const int bb = r >> 8, aa = r & 255;
    const float dx = geom[bb * 3 + 0] - geom[aa * 3 + 0];
    const float dy = geom[bb * 3 + 1] - geom[aa * 3 + 1];
    const float dz = geom[bb * 3 + 2] - geom[aa * 3 + 2];
    sd[tid] = sqrtf(dx * dx + dy * dy + dz * dz + 1e-12f);
  }
  __syncthreads();

  // basis_k tile (zero-padded cols 100..111)
  for (int idx = tid; idx < 32 * HPAD; idx += 256) {
    const int m = idx / HPAD, k = idx % HPAD;
    sA[m][k] = (k < HID) ? cos_bump(sd[m], (float)k * STEP_K, STEP_K) : 0.0f;
  }

  // value-path hidden (contraction length 3 -> scalar is optimal), to global
  for (int idx = tid; idx < 32 * HPAD; idx += 256) {
    const int m = idx / HPAD, h = idx % HPAD;
    float o = 0.0f;
    if (h < HID) {
      float acc = 0.0f;
      for (int t = 0; t < 3; ++t)
        acc += cos_bump(sd[m], (float)t * STEP_V, STEP_V) * vW1[t * HID + h];
      o = swishf(acc * RSQRT3);
    }
    Hv[(size_t)(r0 + m) * HPAD + h] = o;
  }
  __syncthreads();

  // ---- GEMM1: sA(32x100) @ kW1(100x100) -> swish -> sH -------------------
  for (int tile = (wave & 3); tile < NTILES; tile += 4) {
    v8f acc = {};
    const int n = tile * 16 + l16;
    const float* Bp = kW1p + (size_t)(tile * KSTEPS * 32 + lane) * 2;
#pragma unroll 5
    for (int kk = 0; kk < KSTEPS; ++kk) {
      const v2f av = *(const v2f*)&sA[mrow][kk * 4 + 2 * half];
      const v2f bv = *(const v2f*)(Bp + kk * 64);
      acc = wmma_f32(av, bv, acc);
    }
    for (int v = 0; v < 8; ++v)                // C/D layout: row v+8*half
      sH[mt * 16 + v + 8 * half][n] = swishf(0.1f * acc[v]);
  }
  __syncthreads();

  // ---- GEMM2: sH(32x100) @ kW2(100x100) -> swish -> H2 (global) ----------
  for (int tile = (wave & 3); tile < NTILES; tile += 4) {
    v8f acc = {};
    const int n = tile * 16 + l16;
    const float* Bp = kW2p + (size_t)(tile * KSTEPS * 32 + lane) * 2;
#pragma unroll 5
    for (int kk = 0; kk < KSTEPS; ++kk) {
      const v2f av = *(const v2f*)&sH[mrow][kk * 4 + 2 * half];
      const v2f bv = *(const v2f*)(Bp + kk * 64);
      acc = wmma_f32(av, bv, acc);
    }
    for (int v = 0; v < 8; ++v)
      H2[(size_t)(r0 + mt * 16 + v + 8 * half) * HPAD + n] =
          swishf(0.1f * acc[v]);
  }
}

// ---------------------------------------------------------------------------
// Kernel 3: per-b batched contractions (WMMA) + raw attention scores.
//   key_f[:,b,:] = H2[b*256+:, :] @ Wk_eff[b]   (256x100 x 100x32)
//   val_f[:,b,:] = Hv[b*256+:, :] @ Wv_eff[b]
//   scores[a,b]  = key_f[a,b,:] . query[a,:] / 32
// grid = 256*16 (16 a-rows per WG), block = 128 (waves 0/1: key, 2/3: val)
// ---------------------------------------------------------------------------
__global__ void edge_kernel(const float* __restrict__ H2,
                            const float* __restrict__ Hv,
                            const float* __restrict__ Wk_eff,
                            const float* __restrict__ Wv_eff,
                            const float* __restrict__ query,
                            float* __restrict__ val_f,
                            float* __restrict__ scores) {
  __shared__ float keybuf[16][CCH];

  const int wg   = blockIdx.x;
  const int b    = wg >> 4;
  const int a0   = (wg & 15) * 16;
  const int tid  = threadIdx.x;
  const int lane = tid & 31;
  const int wave = tid >> 5;
  const int half = lane >> 4;
  const int l16  = lane & 15;

  const float* A  = (wave < 2) ? H2 : Hv;
  const int ntile = wave & 1;                  // output cols ntile*16 ..
  const float* Bp = ((wave < 2) ? Wk_eff : Wv_eff) + (size_t)b * (HID * CCH) +
                    (size_t)(ntile * KSTEPS * 32 + lane) * 2;
  const size_t rowbase = (size_t)b * NPTS + a0;
  const float* Ar = A + (rowbase + l16) * HPAD + 2 * half;

  v8f acc = {};
  const int n = ntile * 16 + l16;
#pragma unroll 5
  for (int kk = 0; kk < KSTEPS; ++kk) {
    const v2f av = *(const v2f*)(Ar + kk * 4);
    const v2f bv = *(const v2f*)(Bp + kk * 64);
    acc = wmma_f32(av, bv, acc);
  }

  if (wave >= 2) {
    for (int v = 0; v < 8; ++v)
      val_f[(rowbase + v + 8 * half) * CCH + n] = acc[v];
  } else {
    for (int v = 0; v < 8; ++v)
      keybuf[v + 8 * half][n] = acc[v];
  }
  __syncthreads();

  if (tid < 16) {
    const int a = a0 + tid;
    float s = 0.0f;
    for (int i = 0; i < CCH; ++i) s += keybuf[tid][i] * query[a * CCH + i];
    scores[(size_t)a * NPTS + b] = s * (1.0f / (float)CCH);
  }
}

// ---------------------------------------------------------------------------
// Kernel 4: softmax over b + weighted sum of val_f.
// grid = 256 (one per a), block = 256
// ---------------------------------------------------------------------------
__global__ void softmax_out_kernel(const float* __restrict__ scores,
                                   const float* __restrict__ val_f,
                                   float* __restrict__ out) {
  __shared__ float red[NPTS];
  __shared__ float p[NPTS];
  const int a = blockIdx.x;
  const int tid = threadIdx.x;

  const float s = scores[(size_t)a * NPTS + tid];
  red[tid] = s;
  __syncthreads();
  for (int off = 128; off > 0; off >>= 1) {
    if (tid < off) red[tid] = fmaxf(red[tid], red[tid + off]);
    __syncthreads();
  }
  const float mx = red[0];
  __syncthreads();

  const float e = __expf(s - mx);
  red[tid] = e;
  __syncthreads();
  for (int off = 128; off > 0; off >>= 1) {
    if (tid < off) red[tid] += red[tid + off];
    __syncthreads();
  }
  p[tid] = e * __builtin_amdgcn_rcpf(red[0]);
  __syncthreads();

  if (tid < CCH) {
    float acc = 0.0f;
    for (int bb = 0; bb < NPTS; ++bb)
      acc += p[bb] * val_f[((size_t)bb * NPTS + a) * CCH + tid];
    out[a * CCH + tid] = acc;
  }
}

// ---------------------------------------------------------------------------
// Workspace layout (float offsets)
// ---------------------------------------------------------------------------
static const size_t OFF_Q   = 0;                                    //   8192
static const size_t OFF_W1P = OFF_Q   + (size_t)NPTS * CCH;         //  11200
static const size_t OFF_W2P = OFF_W1P + (size_t)WPACK;              //  11200
static const size_t OFF_WK  = OFF_W2P + (size_t)WPACK;              // 819200
static const size_t OFF_WV  = OFF_WK  + (size_t)NPTS * HID * CCH;   // 819200
static const size_t OFF_H2  = OFF_WV  + (size_t)NPTS * HID * CCH;   // 7340032
static const size_t OFF_HV  = OFF_H2  + (size_t)NPTS * NPTS * HPAD; // 7340032
static const size_t OFF_VF  = OFF_HV  + (size_t)NPTS * NPTS * HPAD; // 2097152
static const size_t OFF_SC  = OFF_VF  + (size_t)NPTS * NPTS * CCH;  //   65536

extern "C" void kernel_launch(void* const* d_in, const int* in_sizes, int n_in,
                              void* d_out, int out_size, void* d_ws, size_t ws_size,
                              hipStream_t stream) {
  const float* features = (const float*)d_in[0];  // [1,256,32]
  const float* geometry = (const float*)d_in[1];  // [1,256,3]
  const float* Wq       = (const float*)d_in[2];  // [32,32]
  const float* kW1      = (const float*)d_in[3];  // [100,100]
  const float* kW2      = (const float*)d_in[4];  // [100,100]
  const float* kW3      = (const float*)d_in[5];  // [100,1024]
  const float* vW1      = (const float*)d_in[6];  // [3,100]
  const float* vW2      = (const float*)d_in[7];  // [100,1024]

  float* ws   = (float*)d_ws;
  float* q    = ws + OFF_Q;
  float* w1p  = ws + OFF_W1P;
  float* w2p  = ws + OFF_W2P;
  float* Wk   = ws + OFF_WK;
  float* Wv   = ws + OFF_WV;
  float* H2   = ws + OFF_H2;
  float* Hv   = ws + OFF_HV;
  float* valf = ws + OFF_VF;
  float* sc   = ws + OFF_SC;

  pack_weights_kernel<<<(2 * WPACK + 255) / 256, 256, 0, stream>>>(kW1, kW2,
                                                                   w1p, w2p);
  prep_kernel<<<NPTS, 128, 0, stream>>>(features, Wq, kW3, vW2, q, Wk, Wv);
  hidden_kernel<<<(NPTS * NPTS) / 32, 256, 0, stream>>>(geometry, w1p, w2p,
                                                        vW1, H2, Hv);
  edge_kernel<<<NPTS * (NPTS / 16), 128, 0, stream>>>(H2, Hv, Wk, Wv, q, valf,
                                                      sc);
  softmax_out_kernel<<<NPTS, NPTS, 0, stream>>>(sc, valf, (float*)d_out);
}